// SortAndSelectNeighbours_38285338476781
// MI455X (gfx1250) — compile-verified
//
#include <hip/hip_runtime.h>
#include <stdint.h>

// Problem constants (match the reference)
#define M_NBR 64
#define K_OUT 32
#define RADIUS_F 1.0f
#define BIG_F 1.0e9f
#define ROWS_PER_BLOCK 4
#define BLOCK_THREADS (ROWS_PER_BLOCK * M_NBR) // 256 threads = 8 wave32

__global__ __launch_bounds__(BLOCK_THREADS)
void SortAndSelectNeighbours_kernel(const float* __restrict__ dist,
                                    const int* __restrict__ nidx,
                                    float* __restrict__ odist,
                                    int* __restrict__ oidx,
                                    int nrows) {
    __shared__ float               sD[ROWS_PER_BLOCK][M_NBR];
    __shared__ int                 sI[ROWS_PER_BLOCK][M_NBR];
    __shared__ __align__(16) uint64_t sK[ROWS_PER_BLOCK][M_NBR];

    const int tid = threadIdx.x;
    const int sub = tid >> 6;   // row within block; uniform across each wave32
    const int c   = tid & 63;   // column within row
    const int row = blockIdx.x * ROWS_PER_BLOCK + sub;
    const bool valid = (row < nrows);

    // ---- Stage the row into LDS with CDNA5 async global->LDS copies ----
    if (valid) {
        const float* gd = dist + (size_t)row * M_NBR + c;
        const int*   gi = nidx + (size_t)row * M_NBR + c;
        // Generic pointers to __shared__ carry the workgroup-relative LDS
        // offset in their low 32 bits (shared aperture lives in addr[63:32]).
        uint32_t ldsD = (uint32_t)(uintptr_t)&sD[sub][c];
        uint32_t ldsI = (uint32_t)(uintptr_t)&sI[sub][c];
        uint64_t ga_d = (uint64_t)(uintptr_t)gd;
        uint64_t ga_i = (uint64_t)(uintptr_t)gi;
        asm volatile("global_load_async_to_lds_b32 %0, %1, off"
                     :: "v"(ldsD), "v"(ga_d) : "memory");
        asm volatile("global_load_async_to_lds_b32 %0, %1, off"
                     :: "v"(ldsI), "v"(ga_i) : "memory");
    }
#if __has_builtin(__builtin_amdgcn_s_wait_asynccnt)
    __builtin_amdgcn_s_wait_asynccnt(0);
#else
    asm volatile("s_wait_asynccnt 0" ::: "memory");
#endif
    __syncthreads();

    // ---- Build stable 64-bit sort keys: (f32 bits of tfdist) << 32 | col ----
    // Distances (and the 1e9 sentinel) are positive, so IEEE-754 bit patterns
    // are order-monotonic; the column in the low bits reproduces the stable
    // argsort tie-break exactly (ties only occur at the sentinel).
    float d  = 0.0f;
    int   ix = -1;
    uint64_t skey = ~0ull;
    if (valid) {
        d  = sD[sub][c];
        ix = sI[sub][c];
        float key = (ix < 0) ? BIG_F : d;
        skey = ((uint64_t)__float_as_uint(key) << 32) | (uint32_t)c;
        sK[sub][c] = skey;
    }
    __syncthreads();

    if (!valid) return;

    // ---- Rank = #{j : skey_j < skey_me}; all 64 keys are distinct ----
    // All 32 lanes of a wave share `sub`, so every LDS read below is a
    // same-address broadcast (bank-conflict free). b128 reads cover 2 keys.
    int rank = 0;
    {
        const ulonglong2* k2 = (const ulonglong2*)&sK[sub][0];
#pragma unroll
        for (int j = 0; j < M_NBR / 2; ++j) {
            ulonglong2 p = k2[j];
            rank += (p.x < skey) ? 1 : 0;
            rank += (p.y < skey) ? 1 : 0;
        }
    }

    // ---- Ranks 0..31 each own exactly one output slot ----
    if (rank < K_OUT) {
        // Radius cut uses the ORIGINAL distance (matches reference, including
        // the case where an invalid (-1) neighbour with d<=RADIUS survives).
        const bool beyond = d > RADIUS_F;
        odist[(size_t)row * K_OUT + rank] = beyond ? 0.0f : d;
        oidx [(size_t)row * K_OUT + rank] = beyond ? -1   : ix;
    }
}

extern "C" void kernel_launch(void* const* d_in, const int* in_sizes, int n_in,
                              void* d_out, int out_size, void* d_ws, size_t ws_size,
                              hipStream_t stream) {
    const float* dist = (const float*)d_in[0];
    const int*   nidx = (const int*)d_in[1];
    const int nrows = in_sizes[0] / M_NBR; // 400000

    // d_out = [sdist (nrows*K f32) | snidx (nrows*K i32)] flat, in return order
    float* odist = (float*)d_out;
    int*   oidx  = (int*)(odist + (size_t)nrows * K_OUT);

    const int blocks = (nrows + ROWS_PER_BLOCK - 1) / ROWS_PER_BLOCK;
    hipLaunchKernelGGL(SortAndSelectNeighbours_kernel,
                       dim3(blocks), dim3(BLOCK_THREADS), 0, stream,
                       dist, nidx, odist, oidx, nrows);
}